// Convolution_61383672594926
// MI455X (gfx1250) — compile-verified
//
#include <hip/hip_runtime.h>
#include <math.h>

typedef float v2f __attribute__((ext_vector_type(2)));
typedef float v8f __attribute__((ext_vector_type(8)));
typedef int   v4i __attribute__((ext_vector_type(4)));

// ---- CDNA5 async global->LDS copy (ASYNCcnt-tracked), with safe fallback ----
#if defined(__has_builtin)
#if __has_builtin(__builtin_amdgcn_global_load_async_to_lds_b128) && \
    __has_builtin(__builtin_amdgcn_s_wait_asynccnt)
#define USE_ASYNC_LDS 1
#endif
#endif

#if defined(USE_ASYNC_LDS)
typedef __attribute__((address_space(1))) v4i* gv4i_p;   // global int4*
typedef __attribute__((address_space(3))) v4i* lv4i_p;   // LDS int4*
__device__ __forceinline__ void cp16(float* lds, const float* g) {
  __builtin_amdgcn_global_load_async_to_lds_b128((gv4i_p)(g), (lv4i_p)(lds),
                                                 0, 0);
}
#define ASYNC_FENCE() __builtin_amdgcn_s_wait_asynccnt(0)
#else
__device__ __forceinline__ void cp16(float* lds, const float* g) {
  lds[0] = g[0]; lds[1] = g[1]; lds[2] = g[2]; lds[3] = g[3];
}
#define ASYNC_FENCE()
#endif

namespace {
constexpr int   kN          = 30000;
constexpr int   kE          = 480000;
constexpr int   kMul        = 32;
constexpr float kInvSqrtMul = 0.17677669529663687f;   // 1/sqrt(32)
constexpr float kScNorm     = 0.0625f;                // 1/sqrt(32*8)
constexpr float kInvSqrt8   = 0.35355339059327373f;   // 1/sqrt(8)
constexpr float kOutScale   = 0.03125f;               // 1/sqrt(16) * 1/sqrt(64)
constexpr float kInvSqrt3   = 0.5773502691896258f;
constexpr float kLog2       = 0.6931471805599453f;

// LDS layout for node_stage (floats). Row paddings chosen so every row start
// is 16B-aligned (async B128) AND bank-conflict free on 64 banks:
//   weights stride 40: 2*40 mod 64 = 16 -> the two EXEC halves of a B load
//                      use disjoint bank ranges
//   sS stride 36 / sA stride 12 / sV stride 100: stride*half mod 64 injective
constexpr int kWStride  = 40;
constexpr int kSStride  = 36;
constexpr int kAStride  = 12;
constexpr int kVStride  = 100;
constexpr int kW10Off  = 0;                  // 32 x 40
constexpr int kW11Off  = 1280;               // 32 x 40
constexpr int kWsc0Off = 2560;               // 256 x 40
constexpr int kWsc1Off = 12800;              // 256 x 40
constexpr int kSOff    = 23040;              // 128 x 36
constexpr int kAOff    = 27648;              // 128 x 12
constexpr int kVOff    = 29184;              // 128 x 100
constexpr int kNodeLds = 41984;              // floats = 164 KB of 320 KB WGP LDS
}

// fp32 WMMA: D(16x16) += A(16x4) * B(4x16)  (CDNA5 V_WMMA_F32_16X16X4_F32)
__device__ __forceinline__ v8f wmma_f32(v2f a, v2f b, v8f c) {
  return __builtin_amdgcn_wmma_f32_16x16x4_f32(false, a, false, b, (short)0, c,
                                               false, false);
}

__global__ void __launch_bounds__(256) zero_agg(float* __restrict__ p, int n) {
  int idx = blockIdx.x * blockDim.x + threadIdx.x;
  int stride = gridDim.x * blockDim.x;
  for (int i = idx; i < n; i += stride) p[i] = 0.0f;
}

// ---------------------------------------------------------------------------
// Kernel 1: 8 waves/block, one 16-node tile per wave. Weights + the block's
// 128-node tile staged into LDS with async B128 copies; WMMA fragments fed
// from LDS (padded rows).
//   xbuf[n, 0:32]         = (node_s @ W1_0) * inv_m
//   xbuf[n, 32 + w*3 + d] = (node_v[:,:,d] @ W1_1)[n,w] * inv_m
//   out[n, 0:32]          = sc0   (K=256 outer-product matmul)
//   out[n, 32 + w*3 + d]  = sc1[n,w,d]
// ---------------------------------------------------------------------------
__global__ void __launch_bounds__(256)
node_stage(const float* __restrict__ node_s, const float* __restrict__ node_v,
           const float* __restrict__ node_attr,
           const float* __restrict__ W1_0, const float* __restrict__ W1_1,
           const float* __restrict__ Wsc0, const float* __restrict__ Wsc1,
           float* __restrict__ xbuf, float* __restrict__ out) {
  extern __shared__ float smem[];
  float* sW10  = smem + kW10Off;
  float* sW11  = smem + kW11Off;
  float* sWsc0 = smem + kWsc0Off;
  float* sWsc1 = smem + kWsc1Off;
  float* sS    = smem + kSOff;
  float* sA    = smem + kAOff;
  float* sV    = smem + kVOff;

  const int tid = threadIdx.x;
  const int blockRow0 = blockIdx.x * 128;   // 8 waves * 16 rows

  // ---- async staging (16B chunks; coalesced global reads, once per block) --
  {
    const int r = tid >> 3, c = (tid & 7) * 4;       // W1_*: 256 chunks
    cp16(&sW10[r * kWStride + c], &W1_0[tid * 4]);
    cp16(&sW11[r * kWStride + c], &W1_1[tid * 4]);
  }
  #pragma unroll
  for (int idx = tid; idx < 2048; idx += 256) {      // Wsc*: 2048 chunks
    const int r = idx >> 3, c = (idx & 7) * 4;
    cp16(&sWsc0[r * kWStride + c], &Wsc0[idx * 4]);
    cp16(&sWsc1[r * kWStride + c], &Wsc1[idx * 4]);
  }
  #pragma unroll
  for (int idx = tid; idx < 1024; idx += 256) {      // node_s: 128 x 8 chunks
    const int r = idx >> 3, c = (idx & 7) * 4;
    const int g = blockRow0 + r;
    float* l = &sS[r * kSStride + c];
    if (g < kN) cp16(l, &node_s[g * 32 + c]);
    else { l[0] = 0.f; l[1] = 0.f; l[2] = 0.f; l[3] = 0.f; }
  }
  {
    const int r = tid >> 1, c = (tid & 1) * 4;       // node_attr: 128 x 2
    const int g = blockRow0 + r;
    float* l = &sA[r * kAStride + c];
    if (g < kN) cp16(l, &node_attr[g * 8 + c]);
    else { l[0] = 0.f; l[1] = 0.f; l[2] = 0.f; l[3] = 0.f; }
  }
  #pragma unroll
  for (int idx = tid; idx < 3072; idx += 256) {      // node_v: 128 x 24 chunks
    const int r = idx / 24, c = (idx - r * 24) * 4;
    const int g = blockRow0 + r;
    float* l = &sV[r * kVStride + c];
    if (g < kN) cp16(l, &node_v[g * 96 + c]);
    else { l[0] = 0.f; l[1] = 0.f; l[2] = 0.f; l[3] = 0.f; }
  }
  ASYNC_FENCE();
  __syncthreads();

  const int wave = (blockIdx.x * blockDim.x + tid) >> 5;
  const int ntiles = kN / 16;               // 30000 % 16 == 0
  if (wave >= ntiles) return;               // whole-wave uniform: EXEC full
  const int lane = tid & 31;
  const int half = lane & 15;
  const int hi   = lane >> 4;
  const int row0 = wave * 16;
  const int lrow = (tid >> 5) * 16 + half;  // staged local row for A fragments
  const float* srowS = sS + lrow * kSStride;
  const float* srowA = sA + lrow * kAStride;
  const float* srowV = sV + lrow * kVStride;

  // ---- x0 = node_s @ W1_0 (dual 16-col tiles, shared A fragments) ----
  {
    v8f acc0 = {}, acc1 = {};
    #pragma unroll
    for (int k0 = 0; k0 < kMul; k0 += 4) {
      const int ka = k0 + 2 * hi;
      v2f a, b0, b1;
      a.x  = srowS[ka];
      a.y  = srowS[ka + 1];
      b0.x = sW10[ka * kWStride + half];
      b0.y = sW10[(ka + 1) * kWStride + half];
      b1.x = sW10[ka * kWStride + 16 + half];
      b1.y = sW10[(ka + 1) * kWStride + 16 + half];
      acc0 = wmma_f32(a, b0, acc0);
      acc1 = wmma_f32(a, b1, acc1);
    }
    #pragma unroll
    for (int i = 0; i < 8; ++i) {
      const int r = row0 + i + 8 * hi;
      xbuf[r * 128 + half]      = acc0[i] * kInvSqrtMul;
      xbuf[r * 128 + 16 + half] = acc1[i] * kInvSqrtMul;
    }
  }

  // ---- sc0[n,w] = sum_{uv} s[n,u]*attr[n,v] * Wsc0[(u*8+v),w] ----
  {
    v8f acc0 = {}, acc1 = {};
    #pragma unroll 8
    for (int k0 = 0; k0 < 256; k0 += 4) {
      const int ka = k0 + 2 * hi;
      const int kb = ka + 1;
      v2f a, b0, b1;
      a.x  = srowS[ka >> 3] * srowA[ka & 7];
      a.y  = srowS[kb >> 3] * srowA[kb & 7];
      b0.x = sWsc0[ka * kWStride + half];
      b0.y = sWsc0[kb * kWStride + half];
      b1.x = sWsc0[ka * kWStride + 16 + half];
      b1.y = sWsc0[kb * kWStride + 16 + half];
      acc0 = wmma_f32(a, b0, acc0);
      acc1 = wmma_f32(a, b1, acc1);
    }
    #pragma unroll
    for (int i = 0; i < 8; ++i) {
      const int r = row0 + i + 8 * hi;
      out[r * 128 + half]      = acc0[i] * kScNorm;
      out[r * 128 + 16 + half] = acc1[i] * kScNorm;
    }
  }

  // ---- per vector component d: x1[d] and sc1[d] ----
  for (int d = 0; d < 3; ++d) {
    {
      v8f acc0 = {}, acc1 = {};
      #pragma unroll
      for (int k0 = 0; k0 < kMul; k0 += 4) {
        const int ka = k0 + 2 * hi;
        v2f a, b0, b1;
        a.x  = srowV[ka * 3 + d];
        a.y  = srowV[(ka + 1) * 3 + d];
        b0.x = sW11[ka * kWStride + half];
        b0.y = sW11[(ka + 1) * kWStride + half];
        b1.x = sW11[ka * kWStride + 16 + half];
        b1.y = sW11[(ka + 1) * kWStride + 16 + half];
        acc0 = wmma_f32(a, b0, acc0);
        acc1 = wmma_f32(a, b1, acc1);
      }
      #pragma unroll
      for (int i = 0; i < 8; ++i) {
        const int r = row0 + i + 8 * hi;
        xbuf[r * 128 + 32 + half * 3 + d]        = acc0[i] * kInvSqrtMul;
        xbuf[r * 128 + 32 + (16 + half) * 3 + d] = acc1[i] * kInvSqrtMul;
      }
    }
    {
      v8f acc0 = {}, acc1 = {};
      #pragma unroll 8
      for (int k0 = 0; k0 < 256; k0 += 4) {
        const int ka = k0 + 2 * hi;
        const int kb = ka + 1;
        v2f a, b0, b1;
        a.x  = srowV[(ka >> 3) * 3 + d] * srowA[ka & 7];
        a.y  = srowV[(kb >> 3) * 3 + d] * srowA[kb & 7];
        b0.x = sWsc1[ka * kWStride + half];
        b0.y = sWsc1[kb * kWStride + half];
        b1.x = sWsc1[ka * kWStride + 16 + half];
        b1.y = sWsc1[kb * kWStride + 16 + half];
        acc0 = wmma_f32(a, b0, acc0);
        acc1 = wmma_f32(a, b1, acc1);
      }
      #pragma unroll
      for (int i = 0; i < 8; ++i) {
        const int r = row0 + i + 8 * hi;
        out[r * 128 + 32 + half * 3 + d]        = acc0[i] * kScNorm;
        out[r * 128 + 32 + (16 + half) * 3 + d] = acc1[i] * kScNorm;
      }
    }
  }
}

// ---------------------------------------------------------------------------
// Kernel 2: one wave per edge; lane = channel u. Edge-MLP weights async-staged
// in LDS (shared by 8 edges/block); prefetch of the gathered x row overlaps
// the softplus transcendentals; scatter via global_atomic_add_f32.
// ---------------------------------------------------------------------------
__global__ void __launch_bounds__(256)
edge_stage(const float* __restrict__ edge_attr,
           const float* __restrict__ edge_scalars,
           const float* __restrict__ Wfc1, const float* __restrict__ Wfc2,
           const int* __restrict__ edge_src, const int* __restrict__ edge_dst,
           const float* __restrict__ xbuf, float* __restrict__ aggbuf) {
  __shared__ float sfc1[64];
  __shared__ float sfc2[1024];
  const int tid = threadIdx.x;
  if (tid < 16) cp16(&sfc1[tid * 4], &Wfc1[tid * 4]);
  cp16(&sfc2[tid * 4], &Wfc2[tid * 4]);      // 256 threads x 16B = 4 KB exact
  ASYNC_FENCE();
  __syncthreads();

  const int e = (blockIdx.x * blockDim.x + tid) >> 5;
  const int u = tid & 31;
  if (e >= kE) return;

  const int dst = edge_dst[e];
  const int src = edge_src[e];
  const float* xd = xbuf + (size_t)dst * 128;
  __builtin_prefetch(xd + u, 0, 0);          // global_prefetch: hide gather

  // edge MLP: h = ssp(es @ Wfc1 / sqrt8); w = h @ Wfc2 / sqrt8
  float es[8];
  #pragma unroll
  for (int k = 0; k < 8; ++k) es[k] = edge_scalars[e * 8 + k];
  float h[8];
  #pragma unroll
  for (int j = 0; j < 8; ++j) {
    float acc = 0.0f;
    #pragma unroll
    for (int k = 0; k < 8; ++k) acc = fmaf(es[k], sfc1[k * 8 + j], acc);
    acc *= kInvSqrt8;
    const float sp = fmaxf(acc, 0.0f) + log1pf(expf(-fabsf(acc)));
    h[j] = sp - kLog2;
  }
  float w00 = 0.f, w01 = 0.f, w10 = 0.f, w11 = 0.f;
  #pragma unroll
  for (int k = 0; k < 8; ++k) {
    const float hk = h[k];
    const float* wr = sfc2 + k * 128;
    w00 = fmaf(hk, wr[u], w00);
    w01 = fmaf(hk, wr[32 + u], w01);
    w10 = fmaf(hk, wr[64 + u], w10);
    w11 = fmaf(hk, wr[96 + u], w11);
  }
  w00 *= kInvSqrt8; w01 *= kInvSqrt8; w10 *= kInvSqrt8; w11 *= kInvSqrt8;

  const float sh0 = edge_attr[e * 4 + 0];
  const float s1x = edge_attr[e * 4 + 1];
  const float s1y = edge_attr[e * 4 + 2];
  const float s1z = edge_attr[e * 4 + 3];

  const float x0d = xd[u];
  const float x1x = xd[32 + u * 3 + 0];
  const float x1y = xd[32 + u * 3 + 1];
  const float x1z = xd[32 + u * 3 + 2];

  const float s_a = w00 * x0d * sh0;
  const float dot = x1x * s1x + x1y * s1y + x1z * s1z;
  const float s_b = w11 * dot * kInvSqrt3;
  const float wa  = w01 * x0d;
  const float wb  = w10 * sh0;

  float* ag = aggbuf + (size_t)src * 256;
  atomicAdd(ag + u, s_a);
  atomicAdd(ag + 32 + u, s_b);
  atomicAdd(ag + 64 + u * 3 + 0, wa * s1x);
  atomicAdd(ag + 64 + u * 3 + 1, wa * s1y);
  atomicAdd(ag + 64 + u * 3 + 2, wa * s1z);
  atomicAdd(ag + 64 + (32 + u) * 3 + 0, wb * x1x);
  atomicAdd(ag + 64 + (32 + u) * 3 + 1, wb * x1y);
  atomicAdd(ag + 64 + (32 + u) * 3 + 2, wb * x1z);
}

// ---------------------------------------------------------------------------
// Kernel 3: out0 = agg_s @ W2_0, out1[d] = agg_v[:,:,d] @ W2_1 (K=64),
// W2 async-staged in LDS, accumulated into d_out (which already holds sc).
// ---------------------------------------------------------------------------
__global__ void __launch_bounds__(256)
out_stage(const float* __restrict__ aggbuf, const float* __restrict__ W2_0,
          const float* __restrict__ W2_1, float* __restrict__ out) {
  __shared__ float sW[2 * 64 * kWStride];    // padded rows (20 KB)
  float* sW20 = sW;
  float* sW21 = sW + 64 * kWStride;
  const int tid = threadIdx.x;
  #pragma unroll
  for (int idx = tid; idx < 512; idx += 256) {   // each W2: 512 x 16B chunks
    const int r = idx >> 3, c = (idx & 7) * 4;
    cp16(&sW20[r * kWStride + c], &W2_0[idx * 4]);
    cp16(&sW21[r * kWStride + c], &W2_1[idx * 4]);
  }
  ASYNC_FENCE();
  __syncthreads();

  const int wave = (blockIdx.x * blockDim.x + tid) >> 5;
  const int ntiles = kN / 16;
  if (wave >= ntiles) return;
  const int lane = tid & 31;
  const int half = lane & 15;
  const int hi   = lane >> 4;
  const int row0 = wave * 16;
  const int arow = row0 + half;
  const float* agrow = aggbuf + (size_t)arow * 256;

  // ---- out0 ----
  {
    v8f acc0 = {}, acc1 = {};
    #pragma unroll
    for (int k0 = 0; k0 < 64; k0 += 4) {
      const int ka = k0 + 2 * hi;
      v2f a, b0, b1;
      a.x  = agrow[ka];
      a.y  = agrow[ka + 1];
      b0.x = sW20[ka * kWStride + half];
      b0.y = sW20[(ka + 1) * kWStride + half];
      b1.x = sW20[ka * kWStride + 16 + half];
      b1.y = sW20[(ka + 1) * kWStride + 16 + half];
      acc0 = wmma_f32(a, b0, acc0);
      acc1 = wmma_f32(a, b1, acc1);
    }
    #pragma unroll
    for (int i = 0; i < 8; ++i) {
      const int r = row0 + i + 8 * hi;
      out[r * 128 + half]      += acc0[i] * kOutScale;
      out[r * 128 + 16 + half] += acc1[i] * kOutScale;
    }
  }

  // ---- out1 per vector component d ----
  for (int d = 0; d < 3; ++d) {
    v8f acc0 = {}, acc1 = {};
    #pragma unroll
    for (int k0 = 0; k0 < 64; k0 += 4) {
      const int ka = k0 + 2 * hi;
      v2f a, b0, b1;
      a.x  = agrow[64 + ka * 3 + d];
      a.y  = agrow[64 + (ka + 1) * 3 + d];
      b0.x = sW21[ka * kWStride + half];
      b0.y = sW21[(ka + 1) * kWStride + half];
      b1.x = sW21[ka * kWStride + 16 + half];
      b1.y = sW21[(ka + 1) * kWStride + 16 + half];
      acc0 = wmma_f32(a, b0, acc0);
      acc1 = wmma_f32(a, b1, acc1);
    }
    #pragma unroll
    for (int i = 0; i < 8; ++i) {
      const int r = row0 + i + 8 * hi;
      out[r * 128 + 32 + half * 3 + d]        += acc0[i] * kOutScale;
      out[r * 128 + 32 + (16 + half) * 3 + d] += acc1[i] * kOutScale;
    }
  }
}

extern "C" void kernel_launch(void* const* d_in, const int* in_sizes, int n_in,
                              void* d_out, int out_size, void* d_ws,
                              size_t ws_size, hipStream_t stream) {
  const float* node_s       = (const float*)d_in[0];
  const float* node_v       = (const float*)d_in[1];
  const float* node_attr    = (const float*)d_in[2];
  const float* edge_attr    = (const float*)d_in[3];
  const float* edge_scalars = (const float*)d_in[4];
  const float* W1_0         = (const float*)d_in[5];
  const float* W1_1         = (const float*)d_in[6];
  const float* Wfc1         = (const float*)d_in[7];
  const float* Wfc2         = (const float*)d_in[8];
  const float* W2_0         = (const float*)d_in[9];
  const float* W2_1         = (const float*)d_in[10];
  const float* Wsc0         = (const float*)d_in[11];
  const float* Wsc1         = (const float*)d_in[12];
  const int*   edge_src     = (const int*)d_in[13];
  const int*   edge_dst     = (const int*)d_in[14];
  float* out = (float*)d_out;

  float* xbuf   = (float*)d_ws;                    // (N,128)
  float* aggbuf = xbuf + (size_t)kN * 128;         // (N,256)

  // 0) zero aggregation buffer (required every call for graph replay)
  zero_agg<<<2048, 256, 0, stream>>>(aggbuf, kN * 256);

  // 1) node linears + self-connection (WMMA fp32, async-LDS-staged operands)
  const int ntiles = kN / 16;                      // 1875 waves
  const int nblocks1 = (ntiles + 7) / 8;           // 8 waves / 256-thread block
  node_stage<<<nblocks1, 256, kNodeLds * sizeof(float), stream>>>(
      node_s, node_v, node_attr, W1_0, W1_1, Wsc0, Wsc1, xbuf, out);

  // 2) fused edge MLP + messages + scatter (one wave per edge)
  const int nblocks2 = (kE + 7) / 8;               // 60000 blocks
  edge_stage<<<nblocks2, 256, 0, stream>>>(edge_attr, edge_scalars, Wfc1, Wfc2,
                                           edge_src, edge_dst, xbuf, aggbuf);

  // 3) output linears accumulated onto self-connection (WMMA fp32)
  out_stage<<<nblocks1, 256, 0, stream>>>(aggbuf, W2_0, W2_1, out);
}